// Wrapper_27341761806355
// MI455X (gfx1250) — compile-verified
//
#include <hip/hip_runtime.h>
#include <stdint.h>

// ---------------- problem constants (match reference) ----------------
#define BATCH   8
#define A_NUM   3
#define NCLS    80
#define S0      16384
#define S1      4096
#define S2      1024
#define N_BOXES 64512            // A * (S0+S1+S2)
#define NOFF1   49152            // A*S0
#define NOFF2   61440            // NOFF1 + A*S1
#define E0      (A_NUM*NCLS*S0)  // 3,932,160 elements per batch, level 0
#define E1      (A_NUM*NCLS*S1)  //   983,040
#define E2      (A_NUM*NCLS*S2)  //   245,760
#define TOPK_N  1000
#define MAXOUT  100
#define CONF_T  0.25f
#define IOU_T   0.5f

// ---------------- algorithm constants ----------------
#define NBINS   4096             // histogram bins over logit range [-8, 8)
#define CAP     2048             // candidate capacity per batch (pow2 for bitonic)
#define TILE    1024             // floats per block-tile (256 threads * 4)

// gfx1250 async HBM->LDS path (guarded: falls back to direct loads + prefetch)
#if defined(__HIP_DEVICE_COMPILE__) && defined(__gfx1250__) && \
    __has_builtin(__builtin_amdgcn_global_load_async_to_lds_b128) && \
    __has_builtin(__builtin_amdgcn_s_wait_asynccnt)
#define ASYNC_LDS 1
#else
#define ASYNC_LDS 0
#endif

#if ASYNC_LDS
// Builtin signature (confirmed by hipcc diagnostics): pointers to 16-byte int
// vectors; src in global (AS1), dst in LDS (AS3).
typedef int v4i __attribute__((vector_size(16)));
typedef v4i __attribute__((address_space(1))) gv4i;
typedef v4i __attribute__((address_space(3))) lv4i;
#endif

// ---------------- workspace layout (bytes) ----------------
// [0, 131072)        : u32 hist[BATCH*NBINS]
// [131072, 131104)   : u32 cnt[BATCH]
// [131104, 131136)   : i32 tbin[BATCH]
// [131136, 262208)   : u64 keys[BATCH*CAP]
#define WS_KEYS_OFF  131136
#define WS_INIT_U32  (BATCH * NBINS + BATCH)   // hist + counters must be zeroed per call

__global__ __launch_bounds__(256) void k_init(unsigned int* __restrict__ ws) {
  int i = blockIdx.x * blockDim.x + threadIdx.x;
  if (i < WS_INIT_U32) ws[i] = 0u;
}

// ---- pass 1: per-batch histogram of logits (monotonic proxy for scores) ----
// Double-buffered async HBM->LDS pipeline: tile t+1 streams in via the async
// copy unit (ASYNCcnt) while tile t is binned. Async loads complete in order,
// so s_wait_asynccnt(1) guarantees tile t has landed.
__global__ __launch_bounds__(256) void k_hist(const float* __restrict__ lg0,
                                              const float* __restrict__ lg1,
                                              const float* __restrict__ lg2,
                                              unsigned int* __restrict__ hist) {
  __shared__ unsigned int lh[NBINS];
#if ASYNC_LDS
  __shared__ float tile[2][TILE];
#endif
  const int b = blockIdx.y;
  for (int i = threadIdx.x; i < NBINS; i += 256) lh[i] = 0u;
  __syncthreads();

  const long stride = (long)gridDim.x * TILE;
  for (int lvl = 0; lvl < 3; ++lvl) {
    const float* base;
    long ecnt;
    if (lvl == 0)      { base = lg0 + (size_t)b * E0; ecnt = E0; }
    else if (lvl == 1) { base = lg1 + (size_t)b * E1; ecnt = E1; }
    else               { base = lg2 + (size_t)b * E2; ecnt = E2; }
    // ecnt is a multiple of TILE for every level; tiles are 16B aligned.
    long e0 = (long)blockIdx.x * TILE;
#if ASYNC_LDS
    if (e0 < ecnt) {
      __builtin_amdgcn_global_load_async_to_lds_b128(
          (gv4i*)(uintptr_t)(base + e0 + threadIdx.x * 4),
          (lv4i*)(uintptr_t)&tile[0][threadIdx.x * 4], 0, 0);
    }
    int par = 0;
    for (; e0 < ecnt; e0 += stride) {
      const long e1 = e0 + stride;
      if (e1 < ecnt) {
        __builtin_amdgcn_global_load_async_to_lds_b128(
            (gv4i*)(uintptr_t)(base + e1 + threadIdx.x * 4),
            (lv4i*)(uintptr_t)&tile[par ^ 1][threadIdx.x * 4], 0, 0);
        __builtin_amdgcn_s_wait_asynccnt(1);   // tile t landed; t+1 may fly
      } else {
        __builtin_amdgcn_s_wait_asynccnt(0);   // drain at end of level
      }
      __syncthreads();
      float4 v = *(const float4*)&tile[par][threadIdx.x * 4];
      float xs[4] = {v.x, v.y, v.z, v.w};
#pragma unroll
      for (int q = 0; q < 4; ++q) {
        int bin = (int)floorf((xs[q] + 8.0f) * 256.0f);
        bin = bin < 0 ? 0 : (bin > NBINS - 1 ? NBINS - 1 : bin);
        atomicAdd(&lh[bin], 1u);
      }
      __syncthreads();   // all waves done reading before buffer reuse
      par ^= 1;
    }
#else
    for (; e0 < ecnt; e0 += stride) {
      __builtin_prefetch(base + e0 + stride + threadIdx.x * 4, 0, 0);
      float4 v = *(const float4*)(base + e0 + threadIdx.x * 4);
      float xs[4] = {v.x, v.y, v.z, v.w};
#pragma unroll
      for (int q = 0; q < 4; ++q) {
        int bin = (int)floorf((xs[q] + 8.0f) * 256.0f);
        bin = bin < 0 ? 0 : (bin > NBINS - 1 ? NBINS - 1 : bin);
        atomicAdd(&lh[bin], 1u);
      }
    }
#endif
  }
  __syncthreads();
  for (int i = threadIdx.x; i < NBINS; i += 256)
    if (lh[i]) atomicAdd(&hist[b * NBINS + i], lh[i]);
}

// ---- pass 2: find threshold bin T (suffix count >= TOPK) ----
__global__ void k_thresh(const unsigned int* __restrict__ hist,
                         int* __restrict__ tbin) {
  const int b = blockIdx.x;
  if (threadIdx.x == 0) {
    unsigned long long total = 0;
    int t = 0;
    for (int i = NBINS - 1; i >= 0; --i) {
      total += hist[b * NBINS + i];
      if (total >= TOPK_N) { t = i; break; }
    }
    tbin[b] = t;
  }
}

// ---- pass 3: compact candidates (bin >= T) into 64-bit sort keys ----
// Second streaming pass; logits (~165MB) nearly fit the 192MB L2 warmed by
// pass 1, so keep default RT temporal hints + software prefetch.
__global__ __launch_bounds__(256) void k_compact(const float* __restrict__ lg0,
                                                 const float* __restrict__ lg1,
                                                 const float* __restrict__ lg2,
                                                 const int* __restrict__ tbin,
                                                 unsigned int* __restrict__ cnt,
                                                 unsigned long long* __restrict__ keys) {
  const int b = blockIdx.y;
  const int T = tbin[b];
  for (int lvl = 0; lvl < 3; ++lvl) {
    const float* base;
    int ecnt, sshift, nOff;
    if (lvl == 0)      { base = lg0 + (size_t)b * E0; ecnt = E0; sshift = 14; nOff = 0; }
    else if (lvl == 1) { base = lg1 + (size_t)b * E1; ecnt = E1; sshift = 12; nOff = NOFF1; }
    else               { base = lg2 + (size_t)b * E2; ecnt = E2; sshift = 10; nOff = NOFF2; }
    const int smask = (1 << sshift) - 1;
    for (int e0 = blockIdx.x * TILE; e0 < ecnt; e0 += gridDim.x * TILE) {
      const int e = e0 + threadIdx.x * 4;
      __builtin_prefetch(base + e0 + gridDim.x * TILE + threadIdx.x * 4, 0, 0);
      float4 v = *(const float4*)(base + e);
      const int r  = e >> sshift;       // channel row = a*NCLS + c
      const int sp = e & smask;         // spatial position of element 0
      const int a  = r / NCLS;
      const int c  = r - a * NCLS;
      float xs[4] = {v.x, v.y, v.z, v.w};
#pragma unroll
      for (int q = 0; q < 4; ++q) {
        int bin = (int)floorf((xs[q] + 8.0f) * 256.0f);
        bin = bin < 0 ? 0 : (bin > NBINS - 1 ? NBINS - 1 : bin);
        if (bin >= T) {
          const int n = nOff + (sp + q) * A_NUM + a;              // box index
          const unsigned flat = (unsigned)n * NCLS + (unsigned)c; // flat score index
          unsigned ub = __float_as_uint(xs[q]);                   // order-preserving map
          ub ^= (ub & 0x80000000u) ? 0xFFFFFFFFu : 0x80000000u;
          const unsigned long long key =
              ((unsigned long long)ub << 32) | (unsigned)(~flat); // ties: lower index wins
          const unsigned pos = atomicAdd(&cnt[b], 1u);
          if (pos < CAP) keys[(size_t)b * CAP + pos] = key;
        }
      }
    }
  }
}

// ---- pass 4: per-batch sort + decode + greedy NMS + emit ----
__global__ __launch_bounds__(1024) void k_nms(const float* __restrict__ rg0,
                                              const float* __restrict__ rg1,
                                              const float* __restrict__ rg2,
                                              const float* __restrict__ anchors,
                                              const unsigned int* __restrict__ cnt,
                                              const unsigned long long* __restrict__ keys,
                                              float* __restrict__ out) {
  __shared__ unsigned long long skey[CAP];
  __shared__ float sx1[TOPK_N], sy1[TOPK_N], sx2[TOPK_N], sy2[TOPK_N];
  __shared__ float sar[TOPK_N], ssc[TOPK_N];
  __shared__ int   scl[TOPK_N];
  __shared__ unsigned char ssup[TOPK_N];
  __shared__ int keepIdx[MAXOUT];
  __shared__ int s_nk, s_tk;

  const int b = blockIdx.x;
  const int tid = threadIdx.x;

  int m = (int)cnt[b];
  if (m > CAP) m = CAP;
  for (int i = tid; i < CAP; i += 1024)
    skey[i] = (i < m) ? keys[(size_t)b * CAP + i] : 0ULL;  // 0 sorts below any candidate
  if (tid == 0) { s_nk = 0; s_tk = 0; }
  __syncthreads();

  // Bitonic sort, descending; total-order keys => deterministic output despite
  // nondeterministic atomic-append order.
  for (unsigned k = 2; k <= CAP; k <<= 1) {
    for (unsigned j = k >> 1; j > 0; j >>= 1) {
      for (unsigned i = tid; i < CAP; i += 1024) {
        const unsigned ixj = i ^ j;
        if (ixj > i) {
          const unsigned long long va = skey[i], vb = skey[ixj];
          const bool sw = ((i & k) == 0) ? (va < vb) : (va > vb);
          if (sw) { skey[i] = vb; skey[ixj] = va; }
        }
      }
      __syncthreads();
    }
  }

  // Decode top-TOPK candidates.
  if (tid < TOPK_N) {
    const unsigned long long key = skey[tid];
    if (key != 0ULL) {
      const unsigned flat = ~(unsigned)(key & 0xFFFFFFFFull);
      const int n = (int)(flat / NCLS);
      const int c = (int)(flat - (unsigned)n * NCLS);
      const unsigned hb = (unsigned)(key >> 32);
      const unsigned fb = (hb & 0x80000000u) ? (hb ^ 0x80000000u) : ~hb;
      const float logit = __uint_as_float(fb);
      const float score = 1.0f / (1.0f + expf(-logit));

      const float* rg; int s, off;
      if (n < NOFF1)      { rg = rg0; s = S0; off = 0; }
      else if (n < NOFF2) { rg = rg1; s = S1; off = NOFF1; }
      else                { rg = rg2; s = S2; off = NOFF2; }
      const int loc = n - off;
      const int sp = loc / A_NUM;
      const int a  = loc - sp * A_NUM;
      const float* rb = rg + ((size_t)b * (A_NUM * 4) + a * 4) * s + sp;
      const float dx = rb[0], dy = rb[(size_t)s], dw = rb[2 * (size_t)s], dh = rb[3 * (size_t)s];
      const float* an = anchors + ((size_t)b * N_BOXES + n) * 4;
      const float acx = an[0], acy = an[1], aw = an[2], ah = an[3];
      const float cx = acx + dx * aw, cy = acy + dy * ah;
      const float w = aw * expf(dw), h = ah * expf(dh);
      const float x1 = cx - 0.5f * w, y1 = cy - 0.5f * h;
      const float x2 = cx + 0.5f * w, y2 = cy + 0.5f * h;
      sx1[tid] = x1; sy1[tid] = y1; sx2[tid] = x2; sy2[tid] = y2;
      sar[tid] = fmaxf(x2 - x1, 0.0f) * fmaxf(y2 - y1, 0.0f);
      ssc[tid] = score; scl[tid] = c; ssup[tid] = 0;
    } else {
      sx1[tid] = sy1[tid] = sx2[tid] = sy2[tid] = 0.0f;
      sar[tid] = 0.0f; ssc[tid] = 0.0f; scl[tid] = -1; ssup[tid] = 1;
    }
  }
  __syncthreads();

  // Greedy class-aware NMS, identical semantics to the reference scan.
  for (int i = 0; i < TOPK_N; ++i) {
    const bool valid = (!ssup[i]) && (ssc[i] > CONF_T);
    if (tid == 0 && valid) {
      s_tk++;
      if (s_nk < MAXOUT) keepIdx[s_nk++] = i;
    }
    if (valid) {
      const float xi1 = sx1[i], yi1 = sy1[i], xi2 = sx2[i], yi2 = sy2[i];
      const float ai = sar[i];
      const int ci = scl[i];
      for (int j = i + 1 + tid; j < TOPK_N; j += 1024) {
        if (!ssup[j] && scl[j] == ci) {
          const float xx1 = fmaxf(xi1, sx1[j]);
          const float yy1 = fmaxf(yi1, sy1[j]);
          const float xx2 = fminf(xi2, sx2[j]);
          const float yy2 = fminf(yi2, sy2[j]);
          const float inter = fmaxf(xx2 - xx1, 0.0f) * fmaxf(yy2 - yy1, 0.0f);
          const float uni = ai + sar[j] - inter;
          const float iou = inter / fmaxf(uni, 1e-9f);
          if (iou > IOU_T) ssup[j] = 1;
        }
      }
    }
    __syncthreads();
  }

  // Emit: out[0..B) = num_dets (as float), then dets[B][MAXOUT][6] = {cls, score, x1,y1,x2,y2}.
  if (tid == 0) out[b] = (float)(s_tk < MAXOUT ? s_tk : MAXOUT);
  for (int mm = tid; mm < MAXOUT; mm += 1024) {
    float* d = out + BATCH + ((size_t)b * MAXOUT + mm) * 6;
    if (mm < s_nk) {
      const int i = keepIdx[mm];
      d[0] = (float)scl[i]; d[1] = ssc[i];
      d[2] = sx1[i]; d[3] = sy1[i]; d[4] = sx2[i]; d[5] = sy2[i];
    } else {
      d[0] = -1.0f; d[1] = 0.0f; d[2] = 0.0f; d[3] = 0.0f; d[4] = 0.0f; d[5] = 0.0f;
    }
  }
}

extern "C" void kernel_launch(void* const* d_in, const int* in_sizes, int n_in,
                              void* d_out, int out_size, void* d_ws, size_t ws_size,
                              hipStream_t stream) {
  (void)in_sizes; (void)n_in; (void)out_size; (void)ws_size;
  // setup_inputs() order: logits0, regress0, logits1, regress1, logits2, regress2, anchors
  const float* lg0 = (const float*)d_in[0];
  const float* rg0 = (const float*)d_in[1];
  const float* lg1 = (const float*)d_in[2];
  const float* rg1 = (const float*)d_in[3];
  const float* lg2 = (const float*)d_in[4];
  const float* rg2 = (const float*)d_in[5];
  const float* anchors = (const float*)d_in[6];

  unsigned int* hist = (unsigned int*)d_ws;
  unsigned int* cnt  = hist + BATCH * NBINS;
  int* tbin          = (int*)(cnt + BATCH);
  unsigned long long* keys = (unsigned long long*)((char*)d_ws + WS_KEYS_OFF);
  float* out = (float*)d_out;

  k_init<<<(WS_INIT_U32 + 255) / 256, 256, 0, stream>>>(hist);
  k_hist<<<dim3(256, BATCH), 256, 0, stream>>>(lg0, lg1, lg2, hist);
  k_thresh<<<BATCH, 32, 0, stream>>>(hist, tbin);
  k_compact<<<dim3(256, BATCH), 256, 0, stream>>>(lg0, lg1, lg2, tbin, cnt, keys);
  k_nms<<<BATCH, 1024, 0, stream>>>(rg0, rg1, rg2, anchors, cnt, keys, out);
}